// MambaCipherSolver_19361712570605
// MI455X (gfx1250) — compile-verified
//
#include <hip/hip_runtime.h>
#include <hip/hip_bf16.h>
#include <math.h>

#define D_MODEL   128
#define N_LAYERS  4
#define D_STATE   64
#define D_CONV    4
#define HEADDIM   64
#define D_INNER   256
#define NHEADS    4
#define CONV_DIM  384
#define D_IN_PROJ 644
#define VOCAB_P   64
#define BATCH     8
#define SEQLEN    1024
#define NTOK      (BATCH * SEQLEN)   /* 8192, always a multiple of 16 */
#define EPS_F     1e-5f

typedef float v2f __attribute__((ext_vector_type(2)));
typedef float v8f __attribute__((ext_vector_type(8)));

__device__ __forceinline__ float siluf(float x) { return x / (1.f + expf(-x)); }

// ---------------------------------------------------------------------------
// Embedding gather: x[row, d] = emb[tokens[row], d]
// ---------------------------------------------------------------------------
__global__ void k_embed(const int* __restrict__ tok, const float* __restrict__ emb,
                        float* __restrict__ x) {
  int i = blockIdx.x * blockDim.x + threadIdx.x;
  if (i >= NTOK * D_MODEL) return;
  int row = i >> 7;          // /128
  int d   = i & 127;
  x[i] = emb[tok[row] * D_MODEL + d];
}

// ---------------------------------------------------------------------------
// RMSNorm over rows of length D (blockDim.x == D, one row per block)
// ---------------------------------------------------------------------------
__global__ void k_rmsnorm(const float* __restrict__ x, const float* __restrict__ w,
                          float* __restrict__ out, int D) {
  __shared__ float red[9];
  int row = blockIdx.x;
  float v = x[(size_t)row * D + threadIdx.x];
  float ss = v * v;
  #pragma unroll
  for (int off = 16; off > 0; off >>= 1) ss += __shfl_down(ss, off, 32);
  if ((threadIdx.x & 31) == 0) red[threadIdx.x >> 5] = ss;
  __syncthreads();
  if (threadIdx.x == 0) {
    float t = 0.f;
    int nw = blockDim.x >> 5;
    for (int k = 0; k < nw; k++) t += red[k];
    red[8] = rsqrtf(t / (float)D + EPS_F);
  }
  __syncthreads();
  out[(size_t)row * D + threadIdx.x] = v * red[8] * w[threadIdx.x];
}

// ---------------------------------------------------------------------------
// WMMA f32 GEMM: out(M=NTOK, N) = A(M,K) @ W(N,K)^T  [+ resid(M,N)]
// One wave per 16x16 output tile; K stepped by 4 via V_WMMA_F32_16X16X4_F32.
// N, K are compile-time so the K loop fully unrolls into straight-line
// global_load_b64 + v_wmma with no divergence. Out-of-range weight columns
// are CLAMPED (not zeroed): WMMA column n lives in lane n, and we never
// store columns >= N, so garbage there is harmless.
// Fragment layouts per CDNA5 ISA 7.12.2:
//   A 16x4: lanes 0-15 -> M=lane, K={k0,k0+1}; lanes 16-31 -> M=lane-16, K={k0+2,k0+3}
//   B 4x16: lanes 0-15 -> N=lane, K={k0,k0+1}; lanes 16-31 -> N=lane-16, K={k0+2,k0+3}
//   D 16x16: VGPR r, lanes 0-15 -> (M=r, N=lane); lanes 16-31 -> (M=r+8, N=lane-16)
// ---------------------------------------------------------------------------
template <int N, int K>
__global__ void k_gemm_wmma(const float* __restrict__ A, const float* __restrict__ W,
                            const float* __restrict__ resid, float* __restrict__ out) {
  constexpr int M = NTOK;
  constexpr int tilesN = (N + 15) >> 4;
  constexpr int tilesM = M >> 4;              // M is a multiple of 16
  int gwave = (blockIdx.x * blockDim.x + threadIdx.x) >> 5;
  if (gwave >= tilesM * tilesN) return;
  int lane = threadIdx.x & 31;
  int tm = gwave / tilesN;
  int tn = gwave % tilesN;
  int m0 = tm << 4, n0 = tn << 4;

  int hi  = lane >> 4;        // 0 or 1 -> K sub-pair {0,1} vs {2,3}
  int idx = lane & 15;

  int arow = m0 + idx;        // always < M
  int bcol = n0 + idx;
  if (N & 15) bcol = min(bcol, N - 1);   // clamp: unstored columns only
  const float* Arow = A + (size_t)arow * K + (hi << 1);
  const float* Wrow = W + (size_t)bcol * K + (hi << 1);

  v8f acc = {};
#if __has_builtin(__builtin_amdgcn_wmma_f32_16x16x4_f32)
  #pragma unroll
  for (int k0 = 0; k0 < K; k0 += 4) {
    v2f af = *(const v2f*)(Arow + k0);
    v2f bf = *(const v2f*)(Wrow + k0);
    acc = __builtin_amdgcn_wmma_f32_16x16x4_f32(
        /*neg_a=*/false, af, /*neg_b=*/false, bf,
        /*c_mod=*/(short)0, acc, /*reuse_a=*/false, /*reuse_b=*/false);
  }
#else
  #pragma unroll 4
  for (int k0 = 0; k0 < K; k0++) {
    float bv = W[(size_t)bcol * K + k0];
    #pragma unroll
    for (int r = 0; r < 8; r++)
      acc[r] += A[(size_t)(m0 + r + (hi << 3)) * K + k0] * bv;
  }
#endif

  int ocol = n0 + idx;
  if ((N & 15) == 0 || ocol < N) {
    #pragma unroll
    for (int r = 0; r < 8; r++) {
      int row = m0 + r + (hi << 3);     // always < M
      size_t o = (size_t)row * N + ocol;
      float v = acc[r];
      if (resid) v += resid[o];
      out[o] = v;
    }
  }
}

// ---------------------------------------------------------------------------
// Causal depthwise conv over time on the xBC slice of zxbcdt + bias + SiLU
// zx: (NTOK, 644), xBC = zx[:, 256:640]; out: (NTOK, 384)
// ---------------------------------------------------------------------------
__global__ void k_conv(const float* __restrict__ zx, const float* __restrict__ cw,
                       const float* __restrict__ cb, float* __restrict__ out) {
  int i = blockIdx.x * blockDim.x + threadIdx.x;
  if (i >= NTOK * CONV_DIM) return;
  int c  = i % CONV_DIM;
  int bt = i / CONV_DIM;
  int t  = bt % SEQLEN;
  float acc = cb[c];
  #pragma unroll
  for (int j = 0; j < D_CONV; j++) {
    int tt = t - (D_CONV - 1) + j;
    if (tt >= 0)
      acc += zx[(size_t)(bt - t + tt) * D_IN_PROJ + D_INNER + c] * cw[c * D_CONV + j];
  }
  out[(size_t)bt * CONV_DIM + c] = siluf(acc);
}

// ---------------------------------------------------------------------------
// dt = softplus(dt_raw + dt_bias); dA = exp(dt * (-exp(A_log)))
// ---------------------------------------------------------------------------
__global__ void k_dt(const float* __restrict__ zx, const float* __restrict__ dtb,
                     const float* __restrict__ Alog, float* __restrict__ dt_o,
                     float* __restrict__ dA_o) {
  int i = blockIdx.x * blockDim.x + threadIdx.x;
  if (i >= NTOK * NHEADS) return;
  int hh = i & (NHEADS - 1);
  int row = i >> 2;
  float x = zx[(size_t)row * D_IN_PROJ + (D_INNER + CONV_DIM) + hh] + dtb[hh];
  float sp = (x > 20.f) ? x : log1pf(expf(x));
  float A = -expf(Alog[hh]);
  dt_o[i] = sp;
  dA_o[i] = expf(sp * A);
}

// ---------------------------------------------------------------------------
// Sequential selective-state scan. One block per (b,h); thread = p (headdim).
// Each lane keeps its 64-wide state row in registers; B_t/C_t staged in LDS.
// y[b,t,h,p] = sum_n s[p,n]*C[n] + D[h]*x[b,t,h,p]
// ---------------------------------------------------------------------------
__global__ void k_scan(const float* __restrict__ xc, const float* __restrict__ dts,
                       const float* __restrict__ dAs, const float* __restrict__ Dp,
                       float* __restrict__ y) {
  int bh = blockIdx.x;
  int b  = bh / NHEADS;
  int hh = bh % NHEADS;
  int p  = threadIdx.x;            // 0..63
  __shared__ float Bs[D_STATE];
  __shared__ float Cs[D_STATE];
  float s[D_STATE];
  #pragma unroll
  for (int n = 0; n < D_STATE; n++) s[n] = 0.f;
  float Dph = Dp[hh];

  for (int t = 0; t < SEQLEN; t++) {
    size_t row  = (size_t)b * SEQLEN + t;
    size_t base = row * CONV_DIM;
    Bs[p] = xc[base + D_INNER + p];
    Cs[p] = xc[base + D_INNER + D_STATE + p];
    __syncthreads();
    float x_t  = xc[base + hh * HEADDIM + p];
    size_t hix = row * NHEADS + hh;
    float dA_t = dAs[hix];
    float dx   = dts[hix] * x_t;
    float acc = 0.f;
    #pragma unroll
    for (int n = 0; n < D_STATE; n++) {
      s[n] = dA_t * s[n] + dx * Bs[n];
      acc += s[n] * Cs[n];
    }
    y[row * D_INNER + hh * HEADDIM + p] = acc + Dph * x_t;
    __syncthreads();
  }
}

// ---------------------------------------------------------------------------
// Gated RMSNorm (in place on y): y = rmsnorm(y * silu(z)) * gw, row len 256
// z = zx[:, 0:256] (stride 644). blockDim == 256, one row per block.
// ---------------------------------------------------------------------------
__global__ void k_gatenorm(float* __restrict__ y, const float* __restrict__ zx,
                           const float* __restrict__ gw) {
  __shared__ float red[9];
  int row = blockIdx.x;
  int c   = threadIdx.x;
  float z = zx[(size_t)row * D_IN_PROJ + c];
  float t = y[(size_t)row * D_INNER + c] * siluf(z);
  float ss = t * t;
  #pragma unroll
  for (int off = 16; off > 0; off >>= 1) ss += __shfl_down(ss, off, 32);
  if ((c & 31) == 0) red[c >> 5] = ss;
  __syncthreads();
  if (c == 0) {
    float tot = 0.f;
    for (int k = 0; k < (D_INNER >> 5); k++) tot += red[k];
    red[8] = rsqrtf(tot / (float)D_INNER + EPS_F);
  }
  __syncthreads();
  y[(size_t)row * D_INNER + c] = t * red[8] * gw[c];
}

// ---------------------------------------------------------------------------
template <int N, int K>
static inline void launch_gemm(const float* A, const float* W, const float* resid,
                               float* out, hipStream_t stream) {
  constexpr int tiles = (NTOK >> 4) * ((N + 15) >> 4);
  constexpr int threads = tiles * 32;
  constexpr int blocks = (threads + 255) / 256;
  k_gemm_wmma<N, K><<<blocks, 256, 0, stream>>>(A, W, resid, out);
}

extern "C" void kernel_launch(void* const* d_in, const int* in_sizes, int n_in,
                              void* d_out, int out_size, void* d_ws, size_t ws_size,
                              hipStream_t stream) {
  const int*   tokens     = (const int*)d_in[0];
  const float* emb        = (const float*)d_in[1];
  const float* norm_w     = (const float*)d_in[2];
  const float* in_proj_w  = (const float*)d_in[3];
  const float* conv_w     = (const float*)d_in[4];
  const float* conv_b     = (const float*)d_in[5];
  const float* dt_bias    = (const float*)d_in[6];
  const float* A_log      = (const float*)d_in[7];
  const float* D_param    = (const float*)d_in[8];
  const float* gnorm_w    = (const float*)d_in[9];
  const float* out_proj_w = (const float*)d_in[10];
  const float* norm_f_w   = (const float*)d_in[11];
  const float* lm_head_w  = (const float*)d_in[12];
  float* out = (float*)d_out;

  // Workspace layout (floats). Total ~12.7M floats (~48.4 MB).
  float* ws  = (float*)d_ws;
  float* x   = ws;                                    // (NTOK, 128)
  float* h   = x   + (size_t)NTOK * D_MODEL;          // (NTOK, 128)
  float* zx  = h   + (size_t)NTOK * D_MODEL;          // (NTOK, 644)
  float* xc  = zx  + (size_t)NTOK * D_IN_PROJ;        // (NTOK, 384)
  float* dts = xc  + (size_t)NTOK * CONV_DIM;         // (NTOK, 4)
  float* dAs = dts + (size_t)NTOK * NHEADS;           // (NTOK, 4)
  float* y   = dAs + (size_t)NTOK * NHEADS;           // (NTOK, 256)

  k_embed<<<(NTOK * D_MODEL + 255) / 256, 256, 0, stream>>>(tokens, emb, x);

  for (int i = 0; i < N_LAYERS; i++) {
    k_rmsnorm<<<NTOK, D_MODEL, 0, stream>>>(x, norm_w + i * D_MODEL, h, D_MODEL);
    launch_gemm<D_IN_PROJ, D_MODEL>(h, in_proj_w + (size_t)i * D_IN_PROJ * D_MODEL,
                                    nullptr, zx, stream);
    k_conv<<<(NTOK * CONV_DIM + 255) / 256, 256, 0, stream>>>(
        zx, conv_w + i * CONV_DIM * D_CONV, conv_b + i * CONV_DIM, xc);
    k_dt<<<(NTOK * NHEADS + 255) / 256, 256, 0, stream>>>(
        zx, dt_bias + i * NHEADS, A_log + i * NHEADS, dts, dAs);
    k_scan<<<BATCH * NHEADS, HEADDIM, 0, stream>>>(xc, dts, dAs,
                                                   D_param + i * NHEADS, y);
    k_gatenorm<<<NTOK, D_INNER, 0, stream>>>(y, zx, gnorm_w + i * D_INNER);
    // out_proj + residual, written in place over x (resid == out is elementwise-safe)
    launch_gemm<D_MODEL, D_INNER>(y, out_proj_w + (size_t)i * D_MODEL * D_INNER,
                                  x, x, stream);
  }

  k_rmsnorm<<<NTOK, D_MODEL, 0, stream>>>(x, norm_f_w, h, D_MODEL);
  launch_gemm<VOCAB_P, D_MODEL>(h, lm_head_w, nullptr, out, stream);
}